// trans_GNN_78099685310579
// MI455X (gfx1250) — compile-verified
//
#include <hip/hip_runtime.h>

// ---------------------------------------------------------------------------
// Problem constants (match reference)
// ---------------------------------------------------------------------------
#define FEAT 128
#define EPSBN 1e-5f
#define ROWT 4   // 16-row tiles per block (B-fragment reuse factor)

typedef float v2f __attribute__((ext_vector_type(2)));
typedef float v8f __attribute__((ext_vector_type(8)));

// ---------------------------------------------------------------------------
// fp32 WMMA: accumulate 4 row-tiles x 1 col-tile of A @ W^T, K = 128.
// A-fragment (16x4 f32): lanes 0-15 hold K,K+1 (M=lane); lanes 16-31 K+2,K+3.
// B-fragment (4x16):     N = lane&15, b.x = W[n][K+koff] -> contiguous b64.
// One B load feeds ROWT wmma ops. OOB row tiles clamp loads to nrows-1 so
// EXEC stays all-ones (WMMA requirement); stores are guarded instead.
// ---------------------------------------------------------------------------
template <bool SCALED>
__device__ __forceinline__ void wmma4_k128(const float* __restrict__ A,
                                           const float* __restrict__ W,
                                           int row0, int col0, int nrows,
                                           const float* __restrict__ rowscale,
                                           v8f c[ROWT]) {
  const int lane = threadIdx.x & 31;
  const int m    = lane & 15;
  const int koff = (lane >> 4) << 1;
  const float* __restrict__ wrow = W + (size_t)(col0 + m) * FEAT + koff;
  const float* arow[ROWT];
  float s[ROWT];
#pragma unroll
  for (int t = 0; t < ROWT; ++t) {
    int r = row0 + t * 16 + m;
    r = (r < nrows) ? r : (nrows - 1);           // clamp, keep EXEC uniform
    arow[t] = A + (size_t)r * FEAT + koff;
    s[t]    = SCALED ? rowscale[r] : 1.0f;
  }
#pragma unroll
  for (int k = 0; k < FEAT; k += 4) {
    v2f b;
    b.x = wrow[k];
    b.y = wrow[k + 1];
#pragma unroll
    for (int t = 0; t < ROWT; ++t) {
      v2f a;
      a.x = arow[t][k];
      a.y = arow[t][k + 1];
      if (SCALED) { a.x *= s[t]; a.y *= s[t]; }
      c[t] = __builtin_amdgcn_wmma_f32_16x16x4_f32(false, a, false, b, (short)0,
                                                   c[t], false, false);
    }
  }
}

// ---------------------------------------------------------------------------
// GEMM: OUT[r,c] = act( A0@W0^T (+ A1@W1^T) + bias ),  K = 128 for all inputs.
// Block = 32 * (ncols/16) threads; wave w owns column tile w; block owns
// ROWT row tiles starting at 16*ROWT*blockIdx.x. ncols multiple of 16.
// C/D layout: VGPR j -> M = j (lanes 0-15) / j+8 (lanes 16-31), N = lane&15.
// ---------------------------------------------------------------------------
template <bool SCALED, bool HAS_A1>
__global__ void gemm_bias_act_kernel(const float* __restrict__ A0,
                                     const float* __restrict__ rs0,
                                     const float* __restrict__ W0,
                                     const float* __restrict__ A1,
                                     const float* __restrict__ W1,
                                     const float* __restrict__ bias,
                                     float* __restrict__ OUT,
                                     int nrows, int ncols, int relu) {
  const int wave = threadIdx.x >> 5;
  const int col0 = wave << 4;
  const int row0 = blockIdx.x * (16 * ROWT);
  v8f c[ROWT] = {};
  wmma4_k128<SCALED>(A0, W0, row0, col0, nrows, rs0, c);
  if (HAS_A1) wmma4_k128<false>(A1, W1, row0, col0, nrows, nullptr, c);
  const int lane = threadIdx.x & 31;
  const int n    = lane & 15;
  const int mh   = (lane >> 4) << 3;
  const float bv = bias[col0 + n];
#pragma unroll
  for (int t = 0; t < ROWT; ++t) {
#pragma unroll
    for (int j = 0; j < 8; ++j) {
      int r = row0 + t * 16 + mh + j;
      float v = c[t][j] + bv;
      if (relu) v = fmaxf(v, 0.0f);
      if (r < nrows) OUT[(size_t)r * ncols + col0 + n] = v;
    }
  }
}

// ---------------------------------------------------------------------------
// Degree count: cnt[dst[e]] += 1.0f (hardware f32 atomics into L2)
// ---------------------------------------------------------------------------
__global__ void degree_kernel(const int* __restrict__ dst, float* __restrict__ cnt,
                              int E) {
  int e = blockIdx.x * blockDim.x + threadIdx.x;
  if (e >= E) return;
  __hip_atomic_fetch_add(&cnt[dst[e]], 1.0f, __ATOMIC_RELAXED,
                         __HIP_MEMORY_SCOPE_AGENT);
}

__global__ void invdeg_kernel(const float* __restrict__ cnt,
                              float* __restrict__ invc, int N) {
  int i = blockIdx.x * blockDim.x + threadIdx.x;
  if (i >= N) return;
  invc[i] = 1.0f / fmaxf(cnt[i], 1.0f);
}

// ---------------------------------------------------------------------------
// Edge scatter: one wave32 per edge; lane handles 4 features (float4 gather,
// 4x global_atomic_add_f32 scatter). 51 MB feature matrix lives in 192 MB L2.
// ---------------------------------------------------------------------------
__global__ void scatter_kernel(const int* __restrict__ src,
                               const int* __restrict__ dst,
                               const float* __restrict__ feat,
                               float* __restrict__ agg, int E) {
  int gid  = blockIdx.x * blockDim.x + threadIdx.x;
  int e    = gid >> 5;
  int lane = gid & 31;
  if (e >= E) return;
  int s = src[e], d = dst[e];
  float4 v = *(const float4*)(feat + (size_t)s * FEAT + lane * 4);
  float* p = agg + (size_t)d * FEAT + lane * 4;
  __hip_atomic_fetch_add(p + 0, v.x, __ATOMIC_RELAXED, __HIP_MEMORY_SCOPE_AGENT);
  __hip_atomic_fetch_add(p + 1, v.y, __ATOMIC_RELAXED, __HIP_MEMORY_SCOPE_AGENT);
  __hip_atomic_fetch_add(p + 2, v.z, __ATOMIC_RELAXED, __HIP_MEMORY_SCOPE_AGENT);
  __hip_atomic_fetch_add(p + 3, v.w, __ATOMIC_RELAXED, __HIP_MEMORY_SCOPE_AGENT);
}

// ---------------------------------------------------------------------------
// Gather for layer 2: pull mean-normalized agg2 and h1 rows at node_index.
// One wave per target node, float4 per lane.
// ---------------------------------------------------------------------------
__global__ void gather2_kernel(const int* __restrict__ idx,
                               const float* __restrict__ agg,
                               const float* __restrict__ invc,
                               const float* __restrict__ h1,
                               float* __restrict__ aggg,
                               float* __restrict__ h1g, int M) {
  int gid  = blockIdx.x * blockDim.x + threadIdx.x;
  int m    = gid >> 5;
  int lane = gid & 31;
  if (m >= M) return;
  int node = idx[m];
  float s  = invc[node];
  float4 a = *(const float4*)(agg + (size_t)node * FEAT + lane * 4);
  a.x *= s; a.y *= s; a.z *= s; a.w *= s;
  *(float4*)(aggg + (size_t)m * FEAT + lane * 4) = a;
  float4 h = *(const float4*)(h1 + (size_t)node * FEAT + lane * 4);
  *(float4*)(h1g + (size_t)m * FEAT + lane * 4) = h;
}

// ---------------------------------------------------------------------------
// BatchNorm (training-mode biased stats): one block per column.
// Produces fused scale/shift: y = x*scale + shift.
// ---------------------------------------------------------------------------
__global__ void bn_stats_kernel(const float* __restrict__ X, int nrows, int ncols,
                                const float* __restrict__ g,
                                const float* __restrict__ be,
                                float* __restrict__ scale,
                                float* __restrict__ shift) {
  __shared__ float ssum[256];
  __shared__ float ssq[256];
  const int c = blockIdx.x;
  float sum = 0.0f, sq = 0.0f;
  for (int r = threadIdx.x; r < nrows; r += blockDim.x) {
    float v = X[(size_t)r * ncols + c];
    sum += v;
    sq  += v * v;
  }
  ssum[threadIdx.x] = sum;
  ssq[threadIdx.x]  = sq;
  __syncthreads();
  for (int step = blockDim.x >> 1; step > 0; step >>= 1) {
    if (threadIdx.x < step) {
      ssum[threadIdx.x] += ssum[threadIdx.x + step];
      ssq[threadIdx.x]  += ssq[threadIdx.x + step];
    }
    __syncthreads();
  }
  if (threadIdx.x == 0) {
    float inv_n = 1.0f / (float)nrows;
    float mu    = ssum[0] * inv_n;
    float var   = ssq[0] * inv_n - mu * mu;
    float sc    = g[c] * rsqrtf(var + EPSBN);
    scale[c] = sc;
    shift[c] = be[c] - mu * sc;
  }
}

__global__ void bn_apply_leaky_kernel(float* __restrict__ X,
                                      const float* __restrict__ scale,
                                      const float* __restrict__ shift,
                                      int total, int colmask, float slope) {
  int i = blockIdx.x * blockDim.x + threadIdx.x;
  if (i >= total) return;
  int c = i & colmask;  // ncols is a power of two
  float v = X[i] * scale[c] + shift[c];
  X[i] = (v >= 0.0f) ? v : slope * v;
}

// ---------------------------------------------------------------------------
// Final FC: [M,64] @ [64,1] + b  -> one thread per row.
// ---------------------------------------------------------------------------
__global__ void fc3_kernel(const float* __restrict__ X,
                           const float* __restrict__ w,
                           const float* __restrict__ b,
                           float* __restrict__ out, int M) {
  int m = blockIdx.x * blockDim.x + threadIdx.x;
  if (m >= M) return;
  const float* row = X + (size_t)m * 64;
  float acc = b[0];
#pragma unroll
  for (int k = 0; k < 64; ++k) acc += row[k] * w[k];
  out[m] = acc;
}

// ---------------------------------------------------------------------------
// Host-side launcher
// ---------------------------------------------------------------------------
extern "C" void kernel_launch(void* const* d_in, const int* in_sizes, int n_in,
                              void* d_out, int out_size, void* d_ws, size_t ws_size,
                              hipStream_t stream) {
  const float* x    = (const float*)d_in[0];
  const int*   ei   = (const int*)d_in[1];   // [2, E]
  const int*   nidx = (const int*)d_in[2];   // [M]
  const float* W1l  = (const float*)d_in[3];
  const float* b1   = (const float*)d_in[4];
  const float* W1r  = (const float*)d_in[5];
  const float* W2l  = (const float*)d_in[6];
  const float* b2   = (const float*)d_in[7];
  const float* W2r  = (const float*)d_in[8];
  const float* fcW1 = (const float*)d_in[9];
  const float* fcb1 = (const float*)d_in[10];
  const float* fcW2 = (const float*)d_in[11];
  const float* fcb2 = (const float*)d_in[12];
  const float* fcW3 = (const float*)d_in[13];
  const float* fcb3 = (const float*)d_in[14];
  const float* g1   = (const float*)d_in[15];
  const float* be1  = (const float*)d_in[16];
  const float* g2   = (const float*)d_in[17];
  const float* be2  = (const float*)d_in[18];

  const int N = in_sizes[0] / FEAT;   // 100000
  const int E = in_sizes[1] / 2;      // 1600000
  const int M = in_sizes[2];          // 20000
  const int* src = ei;
  const int* dst = ei + E;

  // workspace carve-out (256B aligned slices)
  char*  base = (char*)d_ws;
  size_t off  = 0;
  auto alloc = [&](size_t bytes) -> void* {
    void* p = base + off;
    off += (bytes + 255) & ~(size_t)255;
    return p;
  };
  float* agg    = (float*)alloc((size_t)N * FEAT * 4);  // reused for both layers
  float* cnt    = (float*)alloc((size_t)N * 4);
  float* invc   = (float*)alloc((size_t)N * 4);
  float* h1     = (float*)alloc((size_t)N * FEAT * 4);
  float* aggg   = (float*)alloc((size_t)M * FEAT * 4);
  float* h1g    = (float*)alloc((size_t)M * FEAT * 4);
  float* h2m    = (float*)alloc((size_t)M * FEAT * 4);
  float* t1     = (float*)alloc((size_t)M * FEAT * 4);
  float* t2     = (float*)alloc((size_t)M * 64 * 4);
  float* scale1 = (float*)alloc(128 * 4);
  float* shift1 = (float*)alloc(128 * 4);
  float* scale2 = (float*)alloc(64 * 4);
  float* shift2 = (float*)alloc(64 * 4);

  const int rowsPerBlk = 16 * ROWT;
  const int gridN = (N + rowsPerBlk - 1) / rowsPerBlk;
  const int gridM = (M + rowsPerBlk - 1) / rowsPerBlk;

  // --- degree / mean normalization ---
  hipMemsetAsync(cnt, 0, (size_t)N * 4, stream);
  hipMemsetAsync(agg, 0, (size_t)N * FEAT * 4, stream);
  degree_kernel<<<(E + 255) / 256, 256, 0, stream>>>(dst, cnt, E);
  invdeg_kernel<<<(N + 255) / 256, 256, 0, stream>>>(cnt, invc, N);

  // --- SAGE layer 1 ---
  scatter_kernel<<<(E * 32 + 255) / 256, 256, 0, stream>>>(src, dst, x, agg, E);
  // h1 = relu( (agg/deg) @ W1l^T + x @ W1r^T + b1 )   [N,128]
  gemm_bias_act_kernel<true, true><<<gridN, 256, 0, stream>>>(
      agg, invc, W1l, x, W1r, b1, h1, N, FEAT, 1);

  // --- SAGE layer 2 (output needed only at node_index rows) ---
  hipMemsetAsync(agg, 0, (size_t)N * FEAT * 4, stream);
  scatter_kernel<<<(E * 32 + 255) / 256, 256, 0, stream>>>(src, dst, h1, agg, E);
  gather2_kernel<<<(M * 32 + 255) / 256, 256, 0, stream>>>(nidx, agg, invc, h1,
                                                           aggg, h1g, M);
  // h2m = aggg @ W2l^T + h1g @ W2r^T + b2    [M,128]
  gemm_bias_act_kernel<false, true><<<gridM, 256, 0, stream>>>(
      aggg, nullptr, W2l, h1g, W2r, b2, h2m, M, FEAT, 0);

  // --- FC1 + BN1 + leaky(0.1) ---
  gemm_bias_act_kernel<false, false><<<gridM, 256, 0, stream>>>(
      h2m, nullptr, fcW1, nullptr, nullptr, fcb1, t1, M, FEAT, 0);
  bn_stats_kernel<<<128, 256, 0, stream>>>(t1, M, 128, g1, be1, scale1, shift1);
  bn_apply_leaky_kernel<<<(M * 128 + 255) / 256, 256, 0, stream>>>(
      t1, scale1, shift1, M * 128, 127, 0.1f);

  // --- FC2 + BN2 + leaky(0.05) ---
  gemm_bias_act_kernel<false, false><<<gridM, 128, 0, stream>>>(
      t1, nullptr, fcW2, nullptr, nullptr, fcb2, t2, M, 64, 0);
  bn_stats_kernel<<<64, 256, 0, stream>>>(t2, M, 64, g2, be2, scale2, shift2);
  bn_apply_leaky_kernel<<<(M * 64 + 255) / 256, 256, 0, stream>>>(
      t2, scale2, shift2, M * 64, 63, 0.05f);

  // --- FC3 -> [M,1] ---
  fc3_kernel<<<(M + 255) / 256, 256, 0, stream>>>(t2, fcW3, fcb3, (float*)d_out, M);
}